// SoftMoELayerWrapper_5042291606065
// MI455X (gfx1250) — compile-verified
//
#include <hip/hip_runtime.h>
#include <hip/hip_bf16.h>
#include <cmath>

// ---------------------------------------------------------------------------
// Soft-MoE layer for MI455X (gfx1250).
// - fp32 V_WMMA_F32_16X16X4_F32 for the per-expert MLP GEMMs (memory-bound on
//   1.2 GB of fp32 weights: ~52us floor at 23.3 TB/s, so fp32 keeps fidelity
//   without changing the bottleneck).
// - Weight tiles streamed global->LDS by the Tensor Data Mover with DOUBLE
//   BUFFERING: chunk c+1's DMA is in flight (TENSORcnt==1 allowed) while the
//   WMMAs consume chunk c from LDS. TDM's pad feature (+16 DW per 64-DW row)
//   makes the B-fragment ds_loads bank-conflict free.
// ---------------------------------------------------------------------------

typedef __attribute__((ext_vector_type(2))) float v2f;
typedef __attribute__((ext_vector_type(8))) float v8f;
typedef __attribute__((ext_vector_type(4))) unsigned int v4u;
typedef __attribute__((ext_vector_type(4))) int v4i;
typedef __attribute__((ext_vector_type(8))) int v8i;

#define MOE_HAVE_TDM __has_builtin(__builtin_amdgcn_tensor_load_to_lds)

namespace moe {

constexpr int B = 32;
constexpr int M = 196;
constexpr int D = 768;
constexpr int NE = 64;   // experts (P = 1 folded away)
constexpr int H = 3072;
constexpr float EPS = 1e-12f;

constexpr int KC = 64;             // K-chunk staged in LDS per pipeline stage
constexpr int AS_STRIDE = KC + 4;  // A tile row stride (floats), conflict-free
constexpr int BS_STRIDE = 80;      // B tile row stride: 64 data + 16 TDM pad
constexpr unsigned AS_FLOATS = 32u * AS_STRIDE;
constexpr unsigned BS_BUF_FLOATS = (unsigned)KC * BS_STRIDE;     // one buffer
constexpr unsigned SMEM_BYTES = (AS_FLOATS + 2u * BS_BUF_FLOATS) * 4u; // 48.5KB
constexpr unsigned BS_OFFSET_BYTES = AS_FLOATS * 4u;

// ---------------- stage 1: L2 normalize tokens ------------------------------
__global__ __launch_bounds__(256) void k_norm_x(const float* __restrict__ x,
                                                float* __restrict__ xn) {
  const int row = blockIdx.x;                 // b*M + m
  const float* xr = x + (size_t)row * D;
  float ss = 0.f;
  for (int i = threadIdx.x; i < D; i += 256) { float v = xr[i]; ss += v * v; }
  __shared__ float red[256];
  red[threadIdx.x] = ss;
  __syncthreads();
  for (int s = 128; s > 0; s >>= 1) {
    if (threadIdx.x < s) red[threadIdx.x] += red[threadIdx.x + s];
    __syncthreads();
  }
  const float inv = 1.0f / fmaxf(sqrtf(red[0]), EPS);
  float* xo = xn + (size_t)row * D;
  for (int i = threadIdx.x; i < D; i += 256) xo[i] = xr[i] * inv;
}

// ---------------- stage 2: normalize expert keys (store transposed) ---------
__global__ __launch_bounds__(256) void k_norm_phi(const float* __restrict__ phi,
                                                  const float* __restrict__ scale,
                                                  float* __restrict__ phinT) {
  const int n = blockIdx.x;                   // expert
  float ss = 0.f;
  for (int d = threadIdx.x; d < D; d += 256) { float v = phi[(size_t)d * NE + n]; ss += v * v; }
  __shared__ float red[256];
  red[threadIdx.x] = ss;
  __syncthreads();
  for (int s = 128; s > 0; s >>= 1) {
    if (threadIdx.x < s) red[threadIdx.x] += red[threadIdx.x + s];
    __syncthreads();
  }
  const float inv = scale[0] / fmaxf(sqrtf(red[0]), EPS);
  for (int d = threadIdx.x; d < D; d += 256)
    phinT[(size_t)n * D + d] = phi[(size_t)d * NE + n] * inv;
}

// ---------------- stage 3: logits[b,m,n] = xn_row . phin_n ------------------
__global__ __launch_bounds__(256) void k_logits(const float* __restrict__ xn,
                                                const float* __restrict__ phinT,
                                                float* __restrict__ logits) {
  const int row = blockIdx.x;                 // b*M + m
  __shared__ float xrow[D];
  const float* xr = xn + (size_t)row * D;
  for (int i = threadIdx.x; i < D; i += 256) xrow[i] = xr[i];
  __syncthreads();
  if (threadIdx.x < NE) {
    const int n = threadIdx.x;
    const float* pr = phinT + (size_t)n * D;
    float s = 0.f;
    #pragma unroll 8
    for (int d = 0; d < D; ++d) s += xrow[d] * pr[d];
    logits[(size_t)row * NE + n] = s;
  }
}

// ---------------- stage 4: dispatch softmax over tokens m -------------------
__global__ __launch_bounds__(256) void k_softmax_d(const float* __restrict__ logits,
                                                   float* __restrict__ dT) {
  const int b = blockIdx.x / NE, n = blockIdx.x % NE;
  const int t = threadIdx.x;
  __shared__ float red[256];
  float v = (t < M) ? logits[((size_t)b * M + t) * NE + n] : -INFINITY;
  red[t] = v;
  __syncthreads();
  for (int s = 128; s > 0; s >>= 1) {
    if (t < s) red[t] = fmaxf(red[t], red[t + s]);
    __syncthreads();
  }
  const float mx = red[0];
  __syncthreads();
  const float e = (t < M) ? expf(v - mx) : 0.f;
  red[t] = e;
  __syncthreads();
  for (int s = 128; s > 0; s >>= 1) {
    if (t < s) red[t] += red[t + s];
    __syncthreads();
  }
  if (t < M) dT[((size_t)b * NE + n) * M + t] = e / red[0];
}

// ---------------- stage 5: combine softmax over experts n -------------------
__global__ __launch_bounds__(64) void k_softmax_c(const float* __restrict__ logits,
                                                  float* __restrict__ c) {
  const int row = blockIdx.x;                 // b*M + m
  const int t = threadIdx.x;                  // expert
  __shared__ float red[64];
  const float v = logits[(size_t)row * NE + t];
  red[t] = v;
  __syncthreads();
  for (int s = 32; s > 0; s >>= 1) {
    if (t < s) red[t] = fmaxf(red[t], red[t + s]);
    __syncthreads();
  }
  const float mx = red[0];
  __syncthreads();
  const float e = expf(v - mx);
  red[t] = e;
  __syncthreads();
  for (int s = 32; s > 0; s >>= 1) {
    if (t < s) red[t] += red[t + s];
    __syncthreads();
  }
  c[(size_t)row * NE + t] = e / red[0];
}

// ---------------- stage 6: slot inputs xs[n,b,:] = sum_m d[b,n,m] xn[b,m,:] -
__global__ __launch_bounds__(256) void k_xs(const float* __restrict__ xn,
                                            const float* __restrict__ dT,
                                            float* __restrict__ xs) {
  const int n = blockIdx.x, b = blockIdx.y;
  const int t = threadIdx.x;
  const float* drow = dT + ((size_t)b * NE + n) * M;
  float a0 = 0.f, a1 = 0.f, a2 = 0.f;
  for (int m = 0; m < M; ++m) {
    const float w = drow[m];
    const float* xr = xn + ((size_t)b * M + m) * D;
    a0 += w * xr[t];
    a1 += w * xr[t + 256];
    a2 += w * xr[t + 512];
  }
  float* o = xs + ((size_t)n * B + b) * D;
  o[t] = a0; o[t + 256] = a1; o[t + 512] = a2;
}

// ---------------- TDM: 2-D tile (tile_d0 x tile_d1) global -> LDS -----------
#if MOE_HAVE_TDM
__device__ __forceinline__ void tdm_load_2d(unsigned lds_off_bytes,
                                            const float* gaddr,
                                            int tensor_d0, int tensor_d1,
                                            int tile_d0, int tile_d1) {
  const unsigned long long ga = (unsigned long long)(uintptr_t)gaddr;
  v4u g0;
  g0[0] = 1u;                                           // count=1, user mode
  g0[1] = lds_off_bytes;                                // lds_addr
  g0[2] = (unsigned)ga;                                 // global_addr[31:0]
  g0[3] = (unsigned)((ga >> 32) & 0x01FFFFFFu) | (2u << 30);  // [56:32], type=2
  v8i g1;
  // data_size=4B (2<<16), pad_enable, pad_interval=64 DW (5), pad_amount=16 DW (15)
  g1[0] = (2 << 16) | (1 << 20) | (5 << 22) | (15 << 25);
  g1[1] = (tensor_d0 & 0xFFFF) << 16;                   // tensor_dim0[15:0]
  g1[2] = ((unsigned)tensor_d0 >> 16) | ((tensor_d1 & 0xFFFF) << 16);
  g1[3] = (int)(((unsigned)tensor_d1 >> 16) | ((unsigned)tile_d0 << 16));
  g1[4] = tile_d1;                                      // tile_dim1 (tile_dim2=0)
  g1[5] = tensor_d0;                                    // dim0 stride lo32
  g1[6] = 0;
  g1[7] = 0;
  const v4i g2 = {0, 0, 0, 0};
  const v4i g3 = {0, 0, 0, 0};
#if defined(__clang_major__) && (__clang_major__ >= 23)
  const v8i g4 = {0, 0, 0, 0, 0, 0, 0, 0};
  __builtin_amdgcn_tensor_load_to_lds(g0, g1, g2, g3, g4, 0);
#else
  __builtin_amdgcn_tensor_load_to_lds(g0, g1, g2, g3, 0);
#endif
}
#endif  // MOE_HAVE_TDM

// ---------------- stages 7/8: per-expert WMMA GEMM (fused bias [+GELU]) -----
// One block = 1 expert slice: 32 output rows x 64 output cols, 8 waves:
// wave = (row-tile mt in {0,1}) x (col-tile ct in 0..3).
// Software pipeline: TDM DMA of weight chunk c+1 overlaps WMMA on chunk c
// (double-buffered B in LDS, s_wait_tensorcnt 1 in steady state).
template <int K, int OUT, bool GELU>
__global__ __launch_bounds__(256) void k_mlp(const float* __restrict__ Abase,
                                             const float* __restrict__ Wbase,
                                             const float* __restrict__ bias,
                                             float* __restrict__ Out) {
  const int n = blockIdx.y;                   // expert
  const int c0blk = blockIdx.x * 64;
  const int tid = threadIdx.x;
  const int lane = tid & 31;
  const int wave = tid >> 5;
  const int mt = wave & 1;                    // row tile (16 rows)
  const int ct = wave >> 1;                   // col tile (16 cols)
  const int col = c0blk + ct * 16 + (lane & 15);

  const float* A = Abase + (size_t)n * 32 * K;      // [32][K]
  const float* W = Wbase + (size_t)n * K * OUT;     // [K][OUT]

  extern __shared__ float smem[];
  float* As = smem;                                 // 32 x AS_STRIDE
  float* Bs = smem + AS_FLOATS;                     // 2 x (KC x BS_STRIDE)

  v8f acc = {};
  const int arow = mt * 16 + (lane & 15);
  const int ksub = (lane >> 4) * 2;                 // 0 (lanes 0-15) / 2 (16-31)
  const int ctcol = ct * 16 + (lane & 15);          // col within 64-wide tile

  constexpr int NCHUNK = K / KC;

#if MOE_HAVE_TDM
  if (wave == 0) {                                  // prologue: chunk 0 -> buf 0
    tdm_load_2d(BS_OFFSET_BYTES, W + c0blk,
                /*tensor_d0=*/OUT, /*tensor_d1=*/K,
                /*tile_d0=*/64, /*tile_d1=*/KC);
  }
#endif

  for (int c = 0; c < NCHUNK; ++c) {
    const int k0 = c * KC;
    const int buf = c & 1;

    // stage A chunk: 32 x KC floats via float4, fully coalesced
    // (As was fully consumed before last iteration's trailing barrier)
    for (int i = tid; i < 32 * (KC / 4); i += 256) {
      const int r = i / (KC / 4), c4 = i % (KC / 4);
      const float4 v = *(const float4*)(A + (size_t)r * K + k0 + c4 * 4);
      *(float4*)(&As[r * AS_STRIDE + c4 * 4]) = v;
    }

#if MOE_HAVE_TDM
    if (wave == 0) {
      if (c + 1 < NCHUNK) {
        // issue DMA for next chunk into the other buffer (read of that buffer
        // finished before the previous trailing barrier), then wait until only
        // that new DMA is outstanding => current chunk's tile has landed.
        tdm_load_2d(BS_OFFSET_BYTES + (unsigned)(1 - buf) * BS_BUF_FLOATS * 4u,
                    W + (size_t)(k0 + KC) * OUT + c0blk,
                    /*tensor_d0=*/OUT, /*tensor_d1=*/K,
                    /*tile_d0=*/64, /*tile_d1=*/KC);
#if __has_builtin(__builtin_amdgcn_s_wait_tensorcnt)
        __builtin_amdgcn_s_wait_tensorcnt(1);
#else
        asm volatile("s_wait_tensorcnt 0x1" ::: "memory");
#endif
      } else {
#if __has_builtin(__builtin_amdgcn_s_wait_tensorcnt)
        __builtin_amdgcn_s_wait_tensorcnt(0);
#else
        asm volatile("s_wait_tensorcnt 0x0" ::: "memory");
#endif
      }
    }
#else
    // fallback: synchronous W tile staging into the same padded layout
    for (int i = tid; i < KC * 16; i += 256) {      // KC rows x 16 float4
      const int r = i / 16, c4 = i % 16;
      const float4 v = *(const float4*)(W + (size_t)(k0 + r) * OUT + c0blk + c4 * 4);
      *(float4*)(&Bs[buf * BS_BUF_FLOATS + r * BS_STRIDE + c4 * 4]) = v;
    }
#endif
    __syncthreads();                                // tiles visible to all waves

    const float* Bc = Bs + buf * BS_BUF_FLOATS;
    #pragma unroll 4
    for (int kk = 0; kk < KC; kk += 4) {
      v2f a;                                        // A frag: 16x4 tile (LDS)
      a.x = As[arow * AS_STRIDE + kk + ksub];
      a.y = As[arow * AS_STRIDE + kk + ksub + 1];
      v2f bf;                                       // B frag: 4x16 tile (LDS)
      bf.x = Bc[(kk + ksub) * BS_STRIDE + ctcol];
      bf.y = Bc[(kk + ksub + 1) * BS_STRIDE + ctcol];
      acc = __builtin_amdgcn_wmma_f32_16x16x4_f32(
          /*neg_a=*/false, a, /*neg_b=*/false, bf,
          /*c_mod=*/(short)0, acc, /*reuse_a=*/false, /*reuse_b=*/false);
    }
    __syncthreads();            // chunk consumed: As and buf free for overwrite
  }

  const float bv = bias[(size_t)n * OUT + col];
  const int rbase = mt * 16 + ((lane & 16) ? 8 : 0);
  #pragma unroll
  for (int r = 0; r < 8; ++r) {
    float v = acc[r] + bv;
    if (GELU) v = 0.5f * v * (1.0f + erff(v * 0.70710678118654752f));
    Out[((size_t)n * 32 + (rbase + r)) * OUT + col] = v;
  }
}

// ---------------- stage 9: y[b,m,:] = sum_n c[b,m,n] * ys[n,b,:] ------------
__global__ __launch_bounds__(256) void k_combine(const float* __restrict__ ys,
                                                 const float* __restrict__ c,
                                                 float* __restrict__ y) {
  const int row = blockIdx.x;                 // b*M + m
  const int b = row / M;
  const int t = threadIdx.x;
  const float* crow = c + (size_t)row * NE;
  float a0 = 0.f, a1 = 0.f, a2 = 0.f;
  for (int n = 0; n < NE; ++n) {
    const float w = crow[n];
    const float* yr = ys + ((size_t)n * B + b) * D;
    a0 += w * yr[t];
    a1 += w * yr[t + 256];
    a2 += w * yr[t + 512];
  }
  float* o = y + (size_t)row * D;
  o[t] = a0; o[t + 256] = a1; o[t + 512] = a2;
}

}  // namespace moe

extern "C" void kernel_launch(void* const* d_in, const int* in_sizes, int n_in,
                              void* d_out, int out_size, void* d_ws, size_t ws_size,
                              hipStream_t stream) {
  using namespace moe;
  (void)in_sizes; (void)n_in; (void)out_size; (void)ws_size;

  const float* x     = (const float*)d_in[0];   // [B,M,D]
  const float* phi   = (const float*)d_in[1];   // [D,NE,1]
  const float* scale = (const float*)d_in[2];   // [1]
  const float* w1    = (const float*)d_in[3];   // [NE,D,H]
  const float* b1    = (const float*)d_in[4];   // [NE,H]
  const float* w2    = (const float*)d_in[5];   // [NE,H,D]
  const float* b2    = (const float*)d_in[6];   // [NE,D]
  float* y = (float*)d_out;                     // [B,M,D]

  // workspace layout (floats)
  float* ws     = (float*)d_ws;
  float* xn     = ws;                            // B*M*D       = 4,816,896
  float* phinT  = xn    + (size_t)B * M * D;     // NE*D        =    49,152
  float* logits = phinT + (size_t)NE * D;        // B*M*NE      =   401,408
  float* dT     = logits+ (size_t)B * M * NE;    // B*NE*M      =   401,408
  float* cmat   = dT    + (size_t)B * NE * M;    // B*M*NE      =   401,408
  float* xs     = cmat  + (size_t)B * M * NE;    // NE*B*D      = 1,572,864
  float* h      = xs    + (size_t)NE * B * D;    // NE*B*H      = 6,291,456
  float* ysbuf  = h     + (size_t)NE * B * H;    // NE*B*D      = 1,572,864

  k_norm_x  <<<B * M, 256, 0, stream>>>(x, xn);
  k_norm_phi<<<NE,    256, 0, stream>>>(phi, scale, phinT);
  k_logits  <<<B * M, 256, 0, stream>>>(xn, phinT, logits);
  k_softmax_d<<<B * NE, 256, 0, stream>>>(logits, dT);
  k_softmax_c<<<B * M,  64, 0, stream>>>(logits, cmat);
  k_xs      <<<dim3(NE, B), 256, 0, stream>>>(xn, dT, xs);

  // Expert MLP: xs[32x768] @ w1 -> gelu -> @ w2, double-buffered TDM + WMMA
  k_mlp<D, H, true ><<<dim3(H / 64, NE), 256, SMEM_BYTES, stream>>>(xs, w1, b1, h);
  k_mlp<H, D, false><<<dim3(D / 64, NE), 256, SMEM_BYTES, stream>>>(h,  w2, b2, ysbuf);

  k_combine<<<B * M, 256, 0, stream>>>(ysbuf, cmat, y);
}